// EinsumBilinear_48773648614214
// MI455X (gfx1250) — compile-verified
//
#include <hip/hip_runtime.h>

typedef __attribute__((ext_vector_type(2))) float v2f;
typedef __attribute__((ext_vector_type(4))) float v4f;
typedef __attribute__((ext_vector_type(8))) float v8f;

#define BDIM 512
#define ADIM 512
#define NDIM 512
#define MDIM 512
#define MBLK 32           // m-block: 8 k-chunks of 4 kept in registers
#define NCHUNK (MBLK / 4) // 8

// out[b,a] = sum_k Aflat[a,k] * G[b,k],  k = n*512 + m
//   Aflat[a,k] = anm[a,n,m]       (HBM stream, A-operand, read once per wave)
//   G[b,k]     = bn[b,n]*bm[b,m]  (on-the-fly B-operand; bm register-blocked)
// Each wave32 owns one 16(a) x 32(b) output strip = 2 WMMA accumulators, so
// every anm load feeds two v_wmma_f32_16x16x4_f32 ops.
__global__ __launch_bounds__(128) void EinsumBilinear_wmma_f32(
    const float* __restrict__ bn,
    const float* __restrict__ anm,
    const float* __restrict__ bm,
    float* __restrict__ out)
{
    const int lane = threadIdx.x & 31;
    const int wave = threadIdx.x >> 5;            // 0..3

    const int a0 = blockIdx.x * 16;               // a-tile shared by waves in WG
    const int b0 = blockIdx.y * 128 + wave * 32;  // two b-tiles: b0, b0+16

    const int lo   = lane & 15;
    const int hi   = lane >> 4;                   // K-pair select per ISA layout
    const int moff = hi * 2;

    const int arow = a0 + lo;                     // A-operand row per lane
    const int bc0  = b0 + lo;                     // B-operand col, tile 0
    const int bc1  = b0 + 16 + lo;                // B-operand col, tile 1

    const float* aRow = anm + (size_t)arow * (NDIM * MDIM) + moff;
    const float* bmB0 = bm  + (size_t)bc0 * MDIM + moff;
    const float* bmB1 = bm  + (size_t)bc1 * MDIM + moff;
    const float* bnB0 = bn  + (size_t)bc0 * NDIM;
    const float* bnB1 = bn  + (size_t)bc1 * NDIM;

    v8f acc0 = {0.f, 0.f, 0.f, 0.f, 0.f, 0.f, 0.f, 0.f};
    v8f acc1 = {0.f, 0.f, 0.f, 0.f, 0.f, 0.f, 0.f, 0.f};

    for (int mb = 0; mb < MDIM; mb += MBLK) {
        // bm for this m-block lives in registers for the whole n loop
        v2f bmr0[NCHUNK], bmr1[NCHUNK];
        #pragma unroll
        for (int c = 0; c < NCHUNK; ++c) {
            bmr0[c] = *(const v2f*)(bmB0 + mb + 4 * c);
            bmr1[c] = *(const v2f*)(bmB1 + mb + 4 * c);
        }

        const float* ap = aRow + mb;
        for (int n = 0; n < NDIM; ++n) {
            const float s0 = bnB0[n];             // cached scalars
            const float s1 = bnB1[n];
            // anm stream is 2KB-strided per lane: prefetch 4 rows ahead
            __builtin_prefetch(ap + 4 * MDIM, 0, 3);
            #pragma unroll
            for (int c = 0; c < NCHUNK; ++c) {
                v2f av = *(const v2f*)(ap + 4 * c);   // one 8B load ...
                v2f bv0, bv1;
                bv0.x = s0 * bmr0[c].x;  bv0.y = s0 * bmr0[c].y;
                bv1.x = s1 * bmr1[c].x;  bv1.y = s1 * bmr1[c].y;
                // ... feeds two independent WMMA chains
                acc0 = __builtin_amdgcn_wmma_f32_16x16x4_f32(
                    false, av, false, bv0, (short)0, acc0, false, false);
                acc1 = __builtin_amdgcn_wmma_f32_16x16x4_f32(
                    false, av, false, bv1, (short)0, acc1, false, false);
            }
            ap += MDIM;
        }
    }

    // D layout: lane holds rows a_local = r + 8*hi (r=0..7), col b_local = lo.
    float* op0 = out + (size_t)bc0 * ADIM + a0 + 8 * hi;
    float* op1 = out + (size_t)bc1 * ADIM + a0 + 8 * hi;
    v4f t0 = {acc0[0], acc0[1], acc0[2], acc0[3]};
    v4f t1 = {acc0[4], acc0[5], acc0[6], acc0[7]};
    v4f t2 = {acc1[0], acc1[1], acc1[2], acc1[3]};
    v4f t3 = {acc1[4], acc1[5], acc1[6], acc1[7]};
    *(v4f*)(op0)     = t0;
    *(v4f*)(op0 + 4) = t1;
    *(v4f*)(op1)     = t2;
    *(v4f*)(op1 + 4) = t3;
}

extern "C" void kernel_launch(void* const* d_in, const int* in_sizes, int n_in,
                              void* d_out, int out_size, void* d_ws, size_t ws_size,
                              hipStream_t stream) {
    const float* bn  = (const float*)d_in[0];  // (B, N)
    const float* anm = (const float*)d_in[1];  // (A, N, M)
    const float* bm  = (const float*)d_in[2];  // (B, M)
    float* out = (float*)d_out;                // (B, A)

    dim3 grid(ADIM / 16, BDIM / 128);          // (32, 4): 4 waves/WG, 2 tiles/wave
    EinsumBilinear_wmma_f32<<<grid, 128, 0, stream>>>(bn, anm, bm, out);
}